// TravelTime_25331717111920
// MI455X (gfx1250) — compile-verified
//
#include <hip/hip_runtime.h>

// ---- problem constants (match reference) ----
#define NRR 512
#define NZZ 256
#define INV_H 2.0f   // 1 / 0.5 km

typedef __attribute__((ext_vector_type(2))) float v2f;
typedef __attribute__((ext_vector_type(8))) float v8f;

// Stage 1: per-pick gather + bilinear eikonal interp + Huber, block-partial loss.
__global__ void tt_main_kernel(
    const int*   __restrict__ station_index,
    const int*   __restrict__ event_index,
    const int*   __restrict__ phase_type,
    const float* __restrict__ phase_time,
    const float* __restrict__ phase_weight,
    const float* __restrict__ event_loc_w,    // [NUM_EVENT,3]
    const float* __restrict__ event_time_w,   // [NUM_EVENT,1]
    const float* __restrict__ station_loc_w,  // [NUM_STATION,3]
    const float* __restrict__ station_dt_w,   // [NUM_STATION,1]
    const float* __restrict__ tt_tables,      // [2,NR,NZ]
    float* __restrict__ pred_out,
    float* __restrict__ resid_out,
    float* __restrict__ partials,
    int n)
{
    const int tid    = threadIdx.x;
    const int stride = blockDim.x * gridDim.x;
    float acc = 0.0f;

    for (int i = blockIdx.x * blockDim.x + tid; i < n; i += stride) {
        // CDNA5 speculative prefetch of next grid-stride iteration's streams
        __builtin_prefetch(&phase_time[i + stride], 0, 0);
        __builtin_prefetch(&event_index[i + stride], 0, 0);
        __builtin_prefetch(&station_index[i + stride], 0, 0);

        const int si = station_index[i];
        const int ei = event_index[i];
        const int pt = phase_type[i];

        const float ex = event_loc_w[ei * 3 + 0];
        const float ey = event_loc_w[ei * 3 + 1];
        const float ez = event_loc_w[ei * 3 + 2];
        const float sx = station_loc_w[si * 3 + 0];
        const float sy = station_loc_w[si * 3 + 1];
        const float sz = station_loc_w[si * 3 + 2];

        const float dx = ex - sx;
        const float dy = ey - sy;
        const float r  = sqrtf(dx * dx + dy * dy);   // epicentral distance
        const float z  = ez - sz;                    // depth difference

        // bilinear interp into tt_tables[pt]
        const float fr   = r * INV_H;                // RG0 = 0
        const float fz   = z * INV_H;                // ZG0 = 0
        const float ir0f = fminf(fmaxf(floorf(fr), 0.0f), (float)(NRR - 2));
        const float iz0f = fminf(fmaxf(floorf(fz), 0.0f), (float)(NZZ - 2));
        const int   ir0  = (int)ir0f;
        const int   iz0  = (int)iz0f;
        const float x    = fr - ir0f;
        const float y    = fz - iz0f;

        const float* __restrict__ T =
            tt_tables + ((size_t)pt * NRR + (size_t)ir0) * NZZ + (size_t)iz0;
        const float Q00 = T[0];
        const float Q01 = T[1];
        const float Q10 = T[NZZ];
        const float Q11 = T[NZZ + 1];

        const float omx = 1.0f - x;
        const float omy = 1.0f - y;
        const float tt  = Q00 * omx * omy + Q01 * omx * y
                        + Q10 * x   * omy + Q11 * x   * y;

        const float pred  = event_time_w[ei] + tt + station_dt_w[si];
        const float resid = phase_time[i] - pred;
        pred_out[i]  = pred;
        resid_out[i] = resid;

        const float d   = fabsf(resid);
        const float hub = (d < 1.0f) ? (0.5f * d * d) : (d - 0.5f);
        acc += hub * phase_weight[i];
    }

    // deterministic block tree-reduction in LDS
    __shared__ float sdata[256];
    sdata[tid] = acc;
    __syncthreads();
    for (int s = 128; s > 0; s >>= 1) {
        if (tid < s) sdata[tid] += sdata[tid + s];
        __syncthreads();
    }
    if (tid == 0) partials[blockIdx.x] = sdata[0];
}

// Stage 2: single block folds all block partials (fixed order -> deterministic).
// Final 32-lane reduction uses V_WMMA_F32_16X16X4_F32 with A = ones(16x4),
// B = (partial per lane, 0): D[i][j] = column-sum of B, so D VGPR0 holds
// partial[j] + partial[j+16] at lane j; 4 shfl_xor steps finish the sum.
__global__ void tt_finalize_kernel(const float* __restrict__ partials,
                                   int nb,
                                   float* __restrict__ loss_out)
{
    const int tid = threadIdx.x;
    float s = 0.0f;
    for (int j = tid; j < nb; j += 256) s += partials[j];

    __shared__ float sdata[256];
    sdata[tid] = s;
    __syncthreads();
    for (int k = 128; k >= 32; k >>= 1) {   // 256 -> 32 partials
        if (tid < k) sdata[tid] += sdata[tid + k];
        __syncthreads();
    }

    if (tid < 32) {   // wave 0 only -> EXEC all-ones within the wave
        const float p = sdata[tid];
        v2f a; a.x = 1.0f; a.y = 1.0f;   // A = ones(16x4)
        v2f b; b.x = p;    b.y = 0.0f;   // B holds the 32 partials + zeros
        v8f c = {};
        v8f d = __builtin_amdgcn_wmma_f32_16x16x4_f32(
            /*neg_a=*/false, a, /*neg_b=*/false, b,
            /*c_mod=*/(short)0, c, /*reuse_a=*/false, /*reuse_b=*/false);
        float v = d[0];                  // lane j: partial[j] + partial[j+16]
        v += __shfl_xor(v, 8, 32);
        v += __shfl_xor(v, 4, 32);
        v += __shfl_xor(v, 2, 32);
        v += __shfl_xor(v, 1, 32);
        if (tid == 0) loss_out[0] = v;
    }
}

extern "C" void kernel_launch(void* const* d_in, const int* in_sizes, int n_in,
                              void* d_out, int out_size, void* d_ws, size_t ws_size,
                              hipStream_t stream)
{
    const int*   station_index = (const int*)  d_in[0];
    const int*   event_index   = (const int*)  d_in[1];
    const int*   phase_type    = (const int*)  d_in[2];
    const float* phase_time    = (const float*)d_in[3];
    const float* phase_weight  = (const float*)d_in[4];
    const float* event_loc_w   = (const float*)d_in[5];
    const float* event_time_w  = (const float*)d_in[6];
    const float* station_loc_w = (const float*)d_in[7];
    const float* station_dt_w  = (const float*)d_in[8];
    const float* tt_tables     = (const float*)d_in[9];
    // d_in[10], d_in[11] (grad_r/grad_z tables) are unused by the reference fwd.

    const int n = in_sizes[0];
    float* out   = (float*)d_out;
    float* pred  = out;             // [0, n)
    float* resid = out + n;         // [n, 2n)
    float* loss  = out + 2 * n;     // [2n]
    float* partials = (float*)d_ws;

    int blocks = 2048;                                  // 16 picks/thread at N=8.4M
    const int maxb_ws = (ws_size >= sizeof(float)) ? (int)(ws_size / sizeof(float)) : 1;
    if (blocks > maxb_ws) blocks = maxb_ws;
    const int nwork = (n + 255) / 256;
    if (blocks > nwork) blocks = nwork;
    if (blocks < 1) blocks = 1;

    tt_main_kernel<<<blocks, 256, 0, stream>>>(
        station_index, event_index, phase_type, phase_time, phase_weight,
        event_loc_w, event_time_w, station_loc_w, station_dt_w, tt_tables,
        pred, resid, partials, n);

    tt_finalize_kernel<<<1, 256, 0, stream>>>(partials, blocks, loss);
}